// DeformConv2d_69277822485122
// MI455X (gfx1250) — compile-verified
//
#include <hip/hip_runtime.h>

typedef float v2f __attribute__((ext_vector_type(2)));
typedef float v8f __attribute__((ext_vector_type(8)));

#define BB    8
#define CIN   64
#define HH    64
#define WW    64
#define COUT  64
#define KK2   9
#define CINK2 576          // CIN * K2
#define NPIX  32           // pixels per block
#define OFFCH 18           // 2 * K2 offset channels
#define HW    4096         // H * W

__global__ __launch_bounds__(256) void deform_conv_fused(
    const float* __restrict__ x,    // (B, CIN, H, W)
    const float* __restrict__ p_w,  // (18, CIN, 3, 3)
    const float* __restrict__ p_b,  // (18,)
    const float* __restrict__ d_w,  // (COUT, CIN, 3, 3)
    const float* __restrict__ d_b,  // (COUT,)
    float* __restrict__ out)        // (B, COUT, H, W)
{
  __shared__ float offs[OFFCH * NPIX];     // offset conv results
  __shared__ int   cpk [KK2 * NPIX];       // packed clamped corner coords
  __shared__ float wts [KK2 * NPIX * 4];   // effective bilinear weights (0 if invalid)
  __shared__ float smp [CIN * NPIX];       // per-tap sampled values [cin][pixel]

  const int tid  = threadIdx.x;
  const int lane = tid & 31;
  const int wv   = tid >> 5;        // wave 0..7
  const int m0   = (wv & 3) * 16;   // cout tile base
  const int n0   = (wv >> 2) * 16;  // pixel tile base (0 or 16)

  const int pix_base = blockIdx.x * NPIX;   // global flat pixel (b*4096 + s)
  const int b      = pix_base >> 12;        // / 4096
  const int s_base = pix_base & (HW - 1);
  const float* xb = x + (size_t)b * (CIN * HW);

  // ---- stage 1: offset conv, 18 channels x 32 pixels (inline, no duplication) ----
  for (int i = tid; i < OFFCH * NPIX; i += 256) {
    const int j  = i >> 5;          // offset channel 0..17
    const int p  = i & (NPIX - 1);
    const int s  = s_base + p;
    const int ho = s >> 6, wo = s & 63;
    float acc = p_b[j];
    for (int kh = 0; kh < 3; ++kh) {
      const int y = ho - 1 + kh;
      if (y < 0 || y >= HH) continue;
      for (int kw = 0; kw < 3; ++kw) {
        const int xc = wo - 1 + kw;
        if (xc < 0 || xc >= WW) continue;
        const float* xp = xb + y * WW + xc;
        const float* wp = p_w + (size_t)j * CIN * 9 + kh * 3 + kw;
        #pragma unroll 8
        for (int c = 0; c < CIN; ++c)
          acc += xp[c * HW] * wp[c * 9];
      }
    }
    offs[i] = acc;
  }
  __syncthreads();

  // ---- stage 2: sampling coords + effective weights per (tap, pixel) ----
  for (int i = tid; i < KK2 * NPIX; i += 256) {
    const int k  = i >> 5;
    const int p  = i & (NPIX - 1);
    const int s  = s_base + p;
    const int ho = s >> 6, wo = s & 63;
    const int kh = k / 3, kw = k % 3;
    const float dy = offs[(2 * k + 0) * NPIX + p];
    const float dx = offs[(2 * k + 1) * NPIX + p];
    const float py = dy + (float)(ho - 1 + kh);
    const float px = dx + (float)(wo - 1 + kw);
    const float y0f = floorf(py), x0f = floorf(px);
    const float ly = py - y0f, lx = px - x0f;
    const int iy0 = (int)y0f, ix0 = (int)x0f;
    const int iy1 = iy0 + 1,  ix1 = ix0 + 1;
    const bool vy0 = (iy0 >= 0) & (iy0 < HH);
    const bool vy1 = (iy1 >= 0) & (iy1 < HH);
    const bool vx0 = (ix0 >= 0) & (ix0 < WW);
    const bool vx1 = (ix1 >= 0) & (ix1 < WW);
    wts[i * 4 + 0] = (vy0 && vx0) ? (1.f - ly) * (1.f - lx) : 0.f;
    wts[i * 4 + 1] = (vy0 && vx1) ? (1.f - ly) * lx         : 0.f;
    wts[i * 4 + 2] = (vy1 && vx0) ? ly * (1.f - lx)         : 0.f;
    wts[i * 4 + 3] = (vy1 && vx1) ? ly * lx                 : 0.f;
    const int cy0 = min(max(iy0, 0), HH - 1), cy1 = min(max(iy1, 0), HH - 1);
    const int cx0 = min(max(ix0, 0), WW - 1), cx1 = min(max(ix1, 0), WW - 1);
    cpk[i] = cy0 | (cy1 << 8) | (cx0 << 16) | (cx1 << 24);
  }
  __syncthreads();

  // ---- stage 3: per-tap bilinear sample into LDS + fp32 WMMA GEMM ----
  v8f acc = {};
  const int hi = lane >> 4;
  const int ln = lane & 15;

  for (int k = 0; k < KK2; ++k) {
    // fill smp[cin][pixel] for this tap (L2-resident gathers)
    for (int i = tid; i < CIN * NPIX; i += 256) {
      const int c  = i >> 5;
      const int p  = i & (NPIX - 1);
      const int ci = k * NPIX + p;
      const int pk  = cpk[ci];
      const int cy0 = pk & 255, cy1 = (pk >> 8) & 255;
      const int cx0 = (pk >> 16) & 255, cx1 = (pk >> 24) & 255;
      const float* xp = xb + c * HW;
      smp[i] = xp[cy0 * WW + cx0] * wts[ci * 4 + 0]
             + xp[cy0 * WW + cx1] * wts[ci * 4 + 1]
             + xp[cy1 * WW + cx0] * wts[ci * 4 + 2]
             + xp[cy1 * WW + cx1] * wts[ci * 4 + 3];
    }
    __syncthreads();

    // 16 WMMAs over cin chunks of 4 (K index within tap = cin)
    #pragma unroll 4
    for (int cc = 0; cc < CIN; cc += 4) {
      const int ca = cc + 2 * hi;            // this half-wave's K pair base
      v2f a, bf;
      // A: d_w[m0+ln, ca+{0,1}, k]  (16x4 fp32 fragment)
      a.x = d_w[(m0 + ln) * CINK2 + (ca + 0) * KK2 + k];
      a.y = d_w[(m0 + ln) * CINK2 + (ca + 1) * KK2 + k];
      // B: smp[ca+{0,1}, n0+ln]     (4x16 fp32 fragment)
      bf.x = smp[(ca + 0) * NPIX + n0 + ln];
      bf.y = smp[(ca + 1) * NPIX + n0 + ln];
      acc = __builtin_amdgcn_wmma_f32_16x16x4_f32(
          /*neg_a=*/false, a, /*neg_b=*/false, bf,
          /*c_mod=*/(short)0, acc, /*reuse_a=*/false, /*reuse_b=*/false);
    }
    __syncthreads();
  }

  // ---- store D tile + bias: lane ln holds N=n0+ln, VGPR v holds M = v + 8*hi ----
  {
    const int sidx = s_base + n0 + ln;
    #pragma unroll
    for (int v = 0; v < 8; ++v) {
      const int o = m0 + hi * 8 + v;
      out[((size_t)b * COUT + o) * HW + sidx] = acc[v] + d_b[o];
    }
  }
}

extern "C" void kernel_launch(void* const* d_in, const int* in_sizes, int n_in,
                              void* d_out, int out_size, void* d_ws, size_t ws_size,
                              hipStream_t stream) {
  (void)in_sizes; (void)n_in; (void)d_ws; (void)ws_size; (void)out_size;
  const float* x   = (const float*)d_in[0];
  const float* p_w = (const float*)d_in[1];
  const float* p_b = (const float*)d_in[2];
  const float* d_w = (const float*)d_in[3];
  const float* d_b = (const float*)d_in[4];
  float* out = (float*)d_out;

  const int npix_total = BB * HW;              // 32768
  dim3 grid(npix_total / NPIX);                // 1024 blocks
  dim3 block(256);                             // 8 wave32
  deform_conv_fused<<<grid, block, 0, stream>>>(x, p_w, p_b, d_w, d_b, out);
}